// ChemConv_70531952935588
// MI455X (gfx1250) — compile-verified
//
#include <hip/hip_runtime.h>
#include <stdint.h>

typedef __attribute__((ext_vector_type(2))) float v2f;
typedef __attribute__((ext_vector_type(8))) float v8f;
typedef __attribute__((ext_vector_type(4))) unsigned int v4u;
typedef __attribute__((ext_vector_type(4))) int v4i;
typedef __attribute__((ext_vector_type(8))) int v8i;

#define NS     32
#define NA     512
#define IND    64
#define OUTD   64
#define FL     12
#define KTOT   (NA * FL)      /* 6144 */
#define LSTR   68             /* LDS pitch (words); 68 mod 64 = 4 -> conflict-free frag reads */
#define NCHUNK (KTOT / 64)    /* 96 */

__device__ __forceinline__ v8f wmma_f32(v2f a, v2f b, v8f c) {
  // D = A(16x4 f32) * B(4x16 f32) + C(16x16 f32)
  return __builtin_amdgcn_wmma_f32_16x16x4_f32(false, a, false, b, (short)0, c, false, false);
}

// ---------------------------------------------------------------------------
// TDM: DMA one 64x64 f32 tile (global row stride = KTOT floats) into LDS,
// with hardware padding of 4 DWORDs after every 64 DWORDs -> 68-word pitch.
// Wave-level op (EXEC ignored); issue from one wave, fence with TENSORcnt.
// Toolchain uses the 6-arg builtin: (g0 v4u, g1 v8i, g2 v4i, g3 v4i, v8i, cpol)
// ---------------------------------------------------------------------------
__device__ __forceinline__ void tdm_load_tile(unsigned lds_off, const float* gsrc) {
  const unsigned long long ga = (unsigned long long)(uintptr_t)gsrc;
  v4u g0;
  g0.x = 1u;                                    // count=1 (valid), user mode
  g0.y = lds_off;                               // lds_addr (byte offset in LDS)
  g0.z = (unsigned)ga;                          // global_addr[31:0]
  g0.w = (unsigned)((ga >> 32) & 0x01FFFFFFu)   // global_addr[56:32]
         | (2u << 30);                          // type = 2 ("image")
  v8i g1;
  g1[0] = (int)((2u << 16)                      // data_size = 4 bytes
              | (1u << 20)                      // pad_enable
              | (5u << 22)                      // pad_interval: 64 DWORDs
              | (3u << 25));                    // pad_amount:   4 DWORDs
  g1[1] = (int)(64u << 16);                     // tensor_dim0 = 64      (bits 79:48)
  g1[2] = (int)(64u << 16);                     // tensor_dim1 = 64      (bits 111:80)
  g1[3] = (int)(64u << 16);                     // tile_dim0   = 64      (bits 127:112)
  g1[4] = 64;                                   // tile_dim1 = 64, tile_dim2 = 0
  g1[5] = (int)KTOT;                            // tensor_dim0_stride = 6144
  g1[6] = 0;                                    // stride hi / dim1_stride lo
  g1[7] = 0;                                    // dim1_stride hi (unused, 2-D)
  const v4i z4 = {0, 0, 0, 0};                  // groups 2/3: 2-D tensor
  const v8i z8 = {0, 0, 0, 0, 0, 0, 0, 0};      // extra operand (6-arg form)
  __builtin_amdgcn_tensor_load_to_lds(g0, g1, z4, z4, z8, 0);
}

// ---------------------------------------------------------------------------
// Kernel 1: W[s][k=n*12+f][o] = sum_d node[s][n][d] * filters[o][f][d]
// ---------------------------------------------------------------------------
__global__ __launch_bounds__(128) void chemconv_build_w(
    const float* __restrict__ node,      // [32][512][64]
    const float* __restrict__ filters,   // [64][12][66]
    float* __restrict__ Wg)              // [32][6144][64]
{
  __shared__ __align__(16) float Ns[64][LSTR];
  __shared__ __align__(16) float Bs[64][LSTR];

  const int s    = blockIdx.x >> 3;
  const int n0   = (blockIdx.x & 7) * 64;
  const int t    = threadIdx.x;
  const int wv   = t >> 5;
  const int lane = t & 31;
  const int half = lane >> 4;
  const int l    = lane & 15;

  { // node tile -> Ns
    const float* src = node + ((size_t)s * NA + n0) * IND;
    const int r0 = t >> 4;
    const int c4 = (t & 15) * 4;
#pragma unroll
    for (int it = 0; it < 8; ++it) {
      const int r = r0 + it * 8;
      *(float4*)&Ns[r][c4] = *(const float4*)(src + (size_t)r * IND + c4);
    }
  }

  for (int f = 0; f < FL; ++f) {
    __syncthreads();
    { // Bs[o][d] = filters[o][f][d]  (rows are 66 floats -> v2f always 8B-aligned)
      const int o0 = t >> 5;
      const int d2 = lane * 2;
#pragma unroll
      for (int it = 0; it < 16; ++it) {
        const int o = o0 + it * 4;
        *(v2f*)&Bs[o][d2] =
            *(const v2f*)(filters + (size_t)o * (FL * 66) + (size_t)f * 66 + d2);
      }
    }
    __syncthreads();

    v8f acc0 = {}, acc1 = {}, acc2 = {}, acc3 = {};
#pragma unroll
    for (int j = 0; j < 16; ++j) {
      const int kk = 4 * j + 2 * half;
      const v2f a  = *(const v2f*)&Ns[16 * wv + l][kk];
      const v2f b0 = *(const v2f*)&Bs[l][kk];
      const v2f b1 = *(const v2f*)&Bs[16 + l][kk];
      const v2f b2 = *(const v2f*)&Bs[32 + l][kk];
      const v2f b3 = *(const v2f*)&Bs[48 + l][kk];
      acc0 = wmma_f32(a, b0, acc0);
      acc1 = wmma_f32(a, b1, acc1);
      acc2 = wmma_f32(a, b2, acc2);
      acc3 = wmma_f32(a, b3, acc3);
    }

#pragma unroll
    for (int r = 0; r < 8; ++r) {
      const int m    = r + 8 * half;
      const int nrow = n0 + 16 * wv + m;
      float* dst = Wg + ((size_t)s * KTOT + (size_t)nrow * FL + f) * OUTD + l;
      dst[0]  = acc0[r];
      dst[16] = acc1[r];
      dst[32] = acc2[r];
      dst[48] = acc3[r];
    }
  }
}

// ---------------------------------------------------------------------------
// Kernel 2: out[s] (512x64) = conn[s] (512x6144) @ W[s] (6144x64) + bond term
// Double-buffered: conn tiles via TDM DMA (TENSORcnt), W via reg-staged LDS
// transpose; one barrier per K-chunk; compute overlaps both transfers.
// ---------------------------------------------------------------------------
__global__ __launch_bounds__(128) void chemconv_main(
    const float* __restrict__ conn,      // [32][512][6144]
    const float* __restrict__ Wg,        // [32][6144][64]
    const float* __restrict__ bond,      // [32][512][12][2]
    const float* __restrict__ filters,   // [64][12][66]
    float* __restrict__ out)             // [32][512][64]
{
  __shared__ __align__(16) float As[2][64][LSTR];   // conn tiles (TDM-filled)
  __shared__ __align__(16) float Wt[2][64][LSTR];   // W chunks transposed [o][k]
  __shared__ float Ft[64][24];                      // filter tail [o][f*2+p]
  __shared__ float Bo[64][24];                      // bond rows   [m][f*2+p]

  const int s    = blockIdx.x >> 3;
  const int a0   = (blockIdx.x & 7) * 64;
  const int t    = threadIdx.x;
  const int wv   = t >> 5;
  const int lane = t & 31;
  const int half = lane >> 4;
  const int l    = lane & 15;

  for (int idx = t; idx < 64 * 24; idx += 128) {
    const int o = idx / 24, j = idx % 24;
    Ft[o][j] = filters[(size_t)o * (FL * 66) + (size_t)(j >> 1) * 66 + 64 + (j & 1)];
  }
  for (int idx = t; idx < 64 * 24; idx += 128) {
    const int m = idx / 24, j = idx % 24;
    Bo[m][j] = bond[((size_t)s * NA + a0 + m) * (FL * 2) + j];
  }

  const float* connS = conn + ((size_t)s * NA + a0) * KTOT;
  const float* wS    = Wg + (size_t)s * KTOT * OUTD;
  const int k0 = t >> 4;
  const int o4 = (t & 15) * 4;

  // ---- prologue: start chunk 0 transfers
  if (wv == 0) tdm_load_tile((unsigned)(uintptr_t)&As[0][0][0], connS);
  float4 wreg[8];
#pragma unroll
  for (int it = 0; it < 8; ++it)
    wreg[it] = *(const float4*)(wS + (size_t)(k0 + it * 8) * OUTD + o4);

  v8f acc0 = {}, acc1 = {}, acc2 = {}, acc3 = {};

  for (int i = 0; i < NCHUNK; ++i) {
    const int cur = i & 1;

    // stage W chunk i into LDS (transposed): buffer last read in iter i-2,
    // ordered by the iter i-1 barrier
#pragma unroll
    for (int it = 0; it < 8; ++it) {
      const int k = k0 + it * 8;
      Wt[cur][o4 + 0][k] = wreg[it].x;
      Wt[cur][o4 + 1][k] = wreg[it].y;
      Wt[cur][o4 + 2][k] = wreg[it].z;
      Wt[cur][o4 + 3][k] = wreg[it].w;
    }
    // start W chunk i+1 global->regs (completes under compute)
    if (i + 1 < NCHUNK) {
#pragma unroll
      for (int it = 0; it < 8; ++it)
        wreg[it] = *(const float4*)(wS + (size_t)((i + 1) * 64 + k0 + it * 8) * OUTD + o4);
    }

    if (wv == 0) __builtin_amdgcn_s_wait_tensorcnt(0);  // conn chunk i landed
    __syncthreads();                                    // publish LDS to all waves

    // start conn chunk i+1 DMA into the other buffer (runs under compute)
    if ((i + 1 < NCHUNK) && wv == 0)
      tdm_load_tile((unsigned)(uintptr_t)&As[cur ^ 1][0][0],
                    connS + (size_t)(i + 1) * 64);

#pragma unroll
    for (int j = 0; j < 16; ++j) {
      const int kk = 4 * j + 2 * half;
      const v2f a  = *(const v2f*)&As[cur][16 * wv + l][kk];
      const v2f b0 = *(const v2f*)&Wt[cur][l][kk];
      const v2f b1 = *(const v2f*)&Wt[cur][16 + l][kk];
      const v2f b2 = *(const v2f*)&Wt[cur][32 + l][kk];
      const v2f b3 = *(const v2f*)&Wt[cur][48 + l][kk];
      acc0 = wmma_f32(a, b0, acc0);
      acc1 = wmma_f32(a, b1, acc1);
      acc2 = wmma_f32(a, b2, acc2);
      acc3 = wmma_f32(a, b3, acc3);
    }
  }

  // epilogue: bond-property term + store (coalesced along o)
#pragma unroll
  for (int r = 0; r < 8; ++r) {
    const int m = 16 * wv + r + 8 * half;
    float s0 = 0.f, s1 = 0.f, s2 = 0.f, s3 = 0.f;
#pragma unroll
    for (int j = 0; j < 24; ++j) {
      const float bv = Bo[m][j];
      s0 += bv * Ft[l][j];
      s1 += bv * Ft[16 + l][j];
      s2 += bv * Ft[32 + l][j];
      s3 += bv * Ft[48 + l][j];
    }
    float* dst = out + ((size_t)s * NA + a0 + m) * OUTD + l;
    dst[0]  = acc0[r] + s0;
    dst[16] = acc1[r] + s1;
    dst[32] = acc2[r] + s2;
    dst[48] = acc3[r] + s3;
  }
}

extern "C" void kernel_launch(void* const* d_in, const int* in_sizes, int n_in,
                              void* d_out, int out_size, void* d_ws, size_t ws_size,
                              hipStream_t stream) {
  const float* node = (const float*)d_in[0];
  const float* conn = (const float*)d_in[1];
  const float* bond = (const float*)d_in[2];
  const float* filt = (const float*)d_in[3];
  float* Wg = (float*)d_ws;   // needs 32*6144*64*4 = 50,331,648 bytes of scratch

  chemconv_build_w<<<dim3(NS * (NA / 64)), dim3(128), 0, stream>>>(node, filt, Wg);
  chemconv_main<<<dim3(NS * (NA / 64)), dim3(128), 0, stream>>>(conn, Wg, bond, filt,
                                                               (float*)d_out);
  (void)in_sizes; (void)n_in; (void)out_size; (void)ws_size;
}